// GAT_3layer_basic_71949292143009
// MI455X (gfx1250) — compile-verified
//
#include <hip/hip_runtime.h>
#include <math.h>

// ---------------------------------------------------------------------------
// GAT 3-layer forward for MI455X (gfx1250, wave32).
// GEMMs use V_WMMA_F32_16X16X4_F32 (fp32 in/out -> matches fp32 reference).
// Edge softmax/aggregation is a bandwidth/atomic streaming pass.
// ---------------------------------------------------------------------------

typedef float v2f __attribute__((ext_vector_type(2)));
typedef float v8f __attribute__((ext_vector_type(8)));

#define DK 256          // K (input feature dim) for every layer
#define KCHUNK 128      // K staging chunk through LDS
#define LDT (KCHUNK + 4)  // padded LDS row stride: 4-bank skew per row
#define BM 32           // block tile rows   (2 wave-rows)
#define BN 64           // block tile cols   (4 wave-cols)

// ------------------------- ordered-float atomic-max helpers ----------------
__device__ __forceinline__ unsigned f2ord(float f) {
  unsigned u = __float_as_uint(f);
  return (u & 0x80000000u) ? ~u : (u | 0x80000000u);
}
__device__ __forceinline__ float ord2f(unsigned u) {
  return __uint_as_float((u & 0x80000000u) ? (u ^ 0x80000000u) : ~u);
}

// ------------------------------- GEMM --------------------------------------
// H[N x Dst] = X[N x 256] * W[256 x Dact]   (columns Dact..Dst-1 are zero)
__global__ __launch_bounds__(256) void gat_gemm_wmma(
    const float* __restrict__ X, const float* __restrict__ W,
    float* __restrict__ H, int N, int Dact, int Dst) {
  __shared__ float As[BM * LDT];   // 32 x 132 f32
  __shared__ float Bst[BN * LDT];  // 64 x 132 f32 (W stored transposed)

  const int tid  = threadIdx.x;
  const int lane = tid & 31;
  const int lo   = lane & 15;
  const int hi   = lane >> 4;          // selects K pair within fragment
  const int wid  = tid >> 5;
  const int wm   = (wid >> 2) * 16;    // wave tile row offset in block
  const int wn   = (wid & 3) * 16;     // wave tile col offset in block
  const int rowBase = blockIdx.x * BM;
  const int colBase = blockIdx.y * BN;

  v8f acc = {};

  for (int kk = 0; kk < DK; kk += KCHUNK) {
    __syncthreads();
    // ---- load A chunk: 32 rows x 128 cols, float4, coalesced ----
    for (int it = 0; it < 4; ++it) {
      int f4  = tid + 256 * it;        // 1024 float4 total
      int r   = f4 >> 5;               // /32 float4 per row
      int c4  = f4 & 31;
      int gr  = rowBase + r;
      float4 v = make_float4(0.f, 0.f, 0.f, 0.f);
      if (gr < N)
        v = *reinterpret_cast<const float4*>(X + (size_t)gr * DK + kk + c4 * 4);
      *reinterpret_cast<float4*>(&As[r * LDT + c4 * 4]) = v;
    }
    // ---- load B chunk transposed: W[kk+k][colBase+n] -> Bst[n][k] ----
    for (int it = 0; it < 32; ++it) {
      int idx = tid + 256 * it;        // 128*64 elements
      int k   = idx >> 6;
      int n   = idx & 63;
      int gc  = colBase + n;
      float v = (gc < Dact) ? W[(size_t)(kk + k) * Dact + gc] : 0.f;
      Bst[n * LDT + k] = v;
    }
    __syncthreads();

    // ---- 32 WMMA steps over this K chunk ----
    const float* arow = &As[(wm + lo) * LDT];
    const float* brow = &Bst[(wn + lo) * LDT];
    #pragma unroll
    for (int k0 = 0; k0 < KCHUNK; k0 += 4) {
      v2f a = *reinterpret_cast<const v2f*>(arow + k0 + 2 * hi);
      v2f b = *reinterpret_cast<const v2f*>(brow + k0 + 2 * hi);
      acc = __builtin_amdgcn_wmma_f32_16x16x4_f32(false, a, false, b,
                                                  (short)0, acc, false, false);
    }
  }

  // ---- store C: VGPR r holds (M = r + 8*hi, N = lo) ----
  const int col = colBase + wn + lo;
  if (col < Dst) {
    #pragma unroll
    for (int r = 0; r < 8; ++r) {
      int row = rowBase + wm + hi * 8 + r;
      if (row < N) H[(size_t)row * Dst + col] = acc[r];
    }
  }
}

// -------------------- fused alpha + per-node init --------------------------
// one wave per node: as = h.a_src, ad = h.a_dst; init m=-inf, denom=0, out=0
__global__ __launch_bounds__(256) void gat_alpha_init(
    const float* __restrict__ H, const float* __restrict__ a_s,
    const float* __restrict__ a_d, float* __restrict__ asv,
    float* __restrict__ adv, unsigned* __restrict__ mv,
    float* __restrict__ dnv, float* __restrict__ outz,
    int N, int Dpad, int Dact) {
  const int lane = threadIdx.x & 31;
  const int i = blockIdx.x * 8 + (threadIdx.x >> 5);
  if (i >= N) return;
  float s1 = 0.f, s2 = 0.f;
  for (int d = lane; d < Dpad; d += 32) {
    float hv = H[(size_t)i * Dpad + d];
    float av = (d < Dact) ? a_s[d] : 0.f;
    float bv = (d < Dact) ? a_d[d] : 0.f;
    s1 += hv * av;
    s2 += hv * bv;
    outz[(size_t)i * Dpad + d] = 0.f;   // zero aggregation target
  }
  #pragma unroll
  for (int off = 16; off > 0; off >>= 1) {
    s1 += __shfl_xor(s1, off, 32);
    s2 += __shfl_xor(s2, off, 32);
  }
  if (lane == 0) {
    asv[i] = s1;
    adv[i] = s2;
    mv[i]  = f2ord(-INFINITY);
    dnv[i] = 0.f;
  }
}

// ----------------- edge pass 1: leaky score + segment max ------------------
__global__ __launch_bounds__(256) void gat_edge_score(
    const int* __restrict__ src, const int* __restrict__ dst,
    const float* __restrict__ asv, const float* __restrict__ adv,
    float* __restrict__ esc, unsigned* __restrict__ mv, int E, int ET) {
  int e = blockIdx.x * blockDim.x + threadIdx.x;
  if (e >= ET) return;
  int s, d;
  if (e < E) { s = src[e]; d = dst[e]; } else { s = d = e - E; }
  float sc = asv[s] + adv[d];
  sc = (sc > 0.f) ? sc : 0.2f * sc;     // leaky_relu, slope 0.2
  esc[e] = sc;
  atomicMax(&mv[d], f2ord(sc));
}

// ----------------- edge pass 2: exp + segment sum (denominator) ------------
__global__ __launch_bounds__(256) void gat_edge_exp(
    const int* __restrict__ dst, const unsigned* __restrict__ mv,
    float* __restrict__ esc, float* __restrict__ dnv, int E, int ET) {
  int e = blockIdx.x * blockDim.x + threadIdx.x;
  if (e >= ET) return;
  int d = (e < E) ? dst[e] : (e - E);
  float p = __expf(esc[e] - ord2f(mv[d]));
  esc[e] = p;
  atomicAdd(&dnv[d], p);
}

// ----------------- edge pass 3: weighted scatter-add of h[src] -------------
// one wave per edge; float4 gathers, f32 global atomic adds (L2-resident)
__global__ __launch_bounds__(256) void gat_aggregate(
    const int* __restrict__ src, const int* __restrict__ dst,
    const float* __restrict__ esc, const float* __restrict__ dnv,
    const float* __restrict__ H, float* __restrict__ OUT,
    int E, int ET, int Dpad) {
  int e = (blockIdx.x * blockDim.x + threadIdx.x) >> 5;
  int lane = threadIdx.x & 31;
  if (e >= ET) return;
  int s, d;
  if (e < E) { s = src[e]; d = dst[e]; } else { s = d = e - E; }
  float coeff = esc[e] / dnv[d];
  const float4* hp = reinterpret_cast<const float4*>(H + (size_t)s * Dpad);
  float* op = OUT + (size_t)d * Dpad;
  int nf4 = Dpad >> 2;
  for (int d4 = lane; d4 < nf4; d4 += 32) {
    float4 hv = hp[d4];
    atomicAdd(op + d4 * 4 + 0, coeff * hv.x);
    atomicAdd(op + d4 * 4 + 1, coeff * hv.y);
    atomicAdd(op + d4 * 4 + 2, coeff * hv.z);
    atomicAdd(op + d4 * 4 + 3, coeff * hv.w);
  }
}

// ----------------- epilogues ----------------------------------------------
__global__ __launch_bounds__(256) void gat_bias_relu(
    float* __restrict__ HIO, const float* __restrict__ b,
    size_t total, int Dpad) {
  size_t t = (size_t)blockIdx.x * blockDim.x + threadIdx.x;
  if (t >= total) return;
  float v = HIO[t] + b[t % Dpad];
  HIO[t] = v > 0.f ? v : 0.f;
}

__global__ __launch_bounds__(256) void gat_bias_out(
    const float* __restrict__ O, const float* __restrict__ b,
    float* __restrict__ out, int N, int Dpad, int Dact) {
  int t = blockIdx.x * blockDim.x + threadIdx.x;
  if (t >= N * Dpad) return;
  int i = t / Dpad, d = t % Dpad;
  if (d < Dact) out[(size_t)i * Dact + d] = O[t] + b[d];
}

// ---------------------------------------------------------------------------
extern "C" void kernel_launch(void* const* d_in, const int* in_sizes, int n_in,
                              void* d_out, int out_size, void* d_ws, size_t ws_size,
                              hipStream_t stream) {
  const float* x  = (const float*)d_in[0];
  const int*   ei = (const int*)d_in[1];
  const float* W1 = (const float*)d_in[2];
  const float* as1 = (const float*)d_in[3];
  const float* ad1 = (const float*)d_in[4];
  const float* b1 = (const float*)d_in[5];
  const float* W2 = (const float*)d_in[6];
  const float* as2 = (const float*)d_in[7];
  const float* ad2 = (const float*)d_in[8];
  const float* b2 = (const float*)d_in[9];
  const float* W3 = (const float*)d_in[10];
  const float* as3 = (const float*)d_in[11];
  const float* ad3 = (const float*)d_in[12];
  const float* b3 = (const float*)d_in[13];

  const int N  = in_sizes[0] / 256;
  const int E  = in_sizes[1] / 2;
  const int ET = E + N;
  const int* src = ei;
  const int* dst = ei + E;

  // workspace layout (floats)
  float* ws = (float*)d_ws;
  float*    hA  = ws;                         // N*256
  float*    hB  = hA + (size_t)N * 256;       // N*256
  float*    h3  = hB + (size_t)N * 256;       // N*16 (layer-3 h, padded)
  float*    o3  = h3 + (size_t)N * 16;        // N*16 (layer-3 aggregate)
  float*    asv = o3 + (size_t)N * 16;        // N
  float*    adv = asv + N;                    // N
  unsigned* mv  = (unsigned*)(adv + N);       // N
  float*    dnv = adv + 2 * (size_t)N;        // N  (right after mv)
  float*    esc = dnv + N;                    // ET

  dim3 blk(256);
  dim3 gemmG((N + BM - 1) / BM, 4);           // 256 output cols
  dim3 gemmG3((N + BM - 1) / BM, 1);          // 16 padded output cols
  int gAlpha = (N + 7) / 8;
  int gEdge  = (ET + 255) / 256;
  int gAgg   = (ET + 7) / 8;                  // one wave per edge
  int gEltF  = (int)(((size_t)N * 256 + 255) / 256);
  int gElt3  = (N * 16 + 255) / 256;

  // -------- layer 1: x -> hA ; aggregate -> hB ; relu(hB+b1) in place ------
  gat_gemm_wmma<<<gemmG, blk, 0, stream>>>(x, W1, hA, N, 256, 256);
  gat_alpha_init<<<gAlpha, blk, 0, stream>>>(hA, as1, ad1, asv, adv, mv, dnv, hB, N, 256, 256);
  gat_edge_score<<<gEdge, blk, 0, stream>>>(src, dst, asv, adv, esc, mv, E, ET);
  gat_edge_exp<<<gEdge, blk, 0, stream>>>(dst, mv, esc, dnv, E, ET);
  gat_aggregate<<<gAgg, blk, 0, stream>>>(src, dst, esc, dnv, hA, hB, E, ET, 256);
  gat_bias_relu<<<gEltF, blk, 0, stream>>>(hB, b1, (size_t)N * 256, 256);

  // -------- layer 2: hB -> hA ; aggregate -> hB ; relu(hB+b2) in place -----
  gat_gemm_wmma<<<gemmG, blk, 0, stream>>>(hB, W2, hA, N, 256, 256);
  gat_alpha_init<<<gAlpha, blk, 0, stream>>>(hA, as2, ad2, asv, adv, mv, dnv, hB, N, 256, 256);
  gat_edge_score<<<gEdge, blk, 0, stream>>>(src, dst, asv, adv, esc, mv, E, ET);
  gat_edge_exp<<<gEdge, blk, 0, stream>>>(dst, mv, esc, dnv, E, ET);
  gat_aggregate<<<gAgg, blk, 0, stream>>>(src, dst, esc, dnv, hA, hB, E, ET, 256);
  gat_bias_relu<<<gEltF, blk, 0, stream>>>(hB, b2, (size_t)N * 256, 256);

  // -------- layer 3: hB -> h3 (10 cols padded to 16) ; aggregate -> o3 -----
  gat_gemm_wmma<<<gemmG3, blk, 0, stream>>>(hB, W3, h3, N, 10, 16);
  gat_alpha_init<<<gAlpha, blk, 0, stream>>>(h3, as3, ad3, asv, adv, mv, dnv, o3, N, 16, 10);
  gat_edge_score<<<gEdge, blk, 0, stream>>>(src, dst, asv, adv, esc, mv, E, ET);
  gat_edge_exp<<<gEdge, blk, 0, stream>>>(dst, mv, esc, dnv, E, ET);
  gat_aggregate<<<gAgg, blk, 0, stream>>>(src, dst, esc, dnv, h3, o3, E, ET, 16);
  gat_bias_out<<<gElt3, blk, 0, stream>>>(o3, b3, (float*)d_out, N, 16, 10);
}